// LIIF_3d_43250320671232
// MI455X (gfx1250) — compile-verified
//
#include <hip/hip_runtime.h>
#include <math.h>

#define H_   256
#define W_   256
#define CIN  64
#define Q_   (H_ * W_)
#define NB   2
#define K0_  577
#define NHID 256
#define NOUT 27
#define W0S  30.0f
#define LSTR 264   // LDS row stride (floats): 256 + 8 to spread banks

typedef __attribute__((ext_vector_type(2))) float v2f;
typedef __attribute__((ext_vector_type(8))) float v8f;

__global__ void zero_f32_kernel(float* p, int n) {
    int i = blockIdx.x * blockDim.x + threadIdx.x;
    if (i < n) p[i] = 0.0f;
}

// Each wave (32 lanes) computes 16 query rows through the whole MLP.
// Block = 64 threads = 2 independent waves (no inter-wave communication).
__global__ __launch_bounds__(64) void liif_mlp_kernel(
    const float* __restrict__ feat, const float* __restrict__ coord,
    const float* __restrict__ w0, const float* __restrict__ b0,
    const float* __restrict__ w1, const float* __restrict__ b1,
    const float* __restrict__ w2, const float* __restrict__ b2,
    const float* __restrict__ w3, const float* __restrict__ b3,
    const float* __restrict__ w4, const float* __restrict__ b4,
    float* __restrict__ out)
{
    __shared__ float lds[2][16][LSTR];

    const int lane = threadIdx.x & 31;
    const int wv   = threadIdx.x >> 5;
    const int half = lane >> 4;      // 0: lanes 0-15, 1: lanes 16-31
    const int nl   = lane & 15;
    float (*hl)[LSTR] = lds[wv];

    const int gw = blockIdx.x * 2 + wv;   // global wave id
    const int m0 = gw * 16;               // first query row of this wave

    // ---- per-lane query metadata; lane L holds metadata for row L%16 ----
    const int   gq = m0 + nl;             // global query index, < NB*Q_
    const int   bq = gq >> 16;            // Q_ == 65536
    float cy = coord[gq * 3 + 0];
    float cx = coord[gq * 3 + 1];
    const float pe = coord[gq * 3 + 2];
    const float eps = 1e-6f;
    cy = fminf(fmaxf(cy, -1.0f + eps), 1.0f - eps);
    cx = fminf(fmaxf(cx, -1.0f + eps), 1.0f - eps);
    int iy = (int)rintf((cy + 1.0f) * (0.5f * H_) - 0.5f);
    int ix = (int)rintf((cx + 1.0f) * (0.5f * W_) - 0.5f);
    iy = min(max(iy, 0), H_ - 1);
    ix = min(max(ix, 0), W_ - 1);

    // Branchless 3x3 neighborhood descriptors (per lane):
    // clamped row offsets (in floats) and col offsets, plus 3-bit validity.
    const int yoff0 = max(iy - 1, 0) * W_;
    const int yoff1 = iy * W_;
    const int yoff2 = min(iy + 1, H_ - 1) * W_;
    const int xoff0 = max(ix - 1, 0);
    const int xoff1 = ix;
    const int xoff2 = min(ix + 1, W_ - 1);
    const int vy = 2 | (iy > 0 ? 1 : 0) | (iy < H_ - 1 ? 4 : 0);
    const int vx = 2 | (ix > 0 ? 1 : 0) | (ix < W_ - 1 ? 4 : 0);

    // Metadata redistributed to D-fragment layout (m = v + 8*half): needed for
    // the pe rank-1 update (qpe) and the fold3 scatter (qy/qx/qb). Done while
    // all lanes are active.
    int   qy[8], qx[8], qb[8];
    float qpe[8];
#pragma unroll
    for (int v = 0; v < 8; ++v) {
        const int mm = v + 8 * half;
        qy[v]  = __shfl(iy, mm, 32);
        qx[v]  = __shfl(ix, mm, 32);
        qb[v]  = __shfl(bq, mm, 32);
        qpe[v] = __shfl(pe, mm, 32);
    }

    const v8f zero8 = {0.f, 0.f, 0.f, 0.f, 0.f, 0.f, 0.f, 0.f};
    v8f acc[16];
#pragma unroll
    for (int t = 0; t < 16; ++t) acc[t] = zero8;

    // =======================  layer 0: K = 577  ==================================
    // WMMA loop covers K in [0,576): pure feat gather, fully branchless.
    // A element for (row nl, K): K = c*9 + r, r = 3*di + dj -> feat[c, iy+di-1, ix+dj-1].
    // The K=576 pe column is a rank-1 update folded into the activation step.
    const float* featb = feat + (size_t)bq * (CIN * H_ * W_);
    // per-v incremental (c, r): K_v = k0 + v + 2*half, starts at v + 2*half < 9.
    int c0 = 0, r0 = 2 * half;          // v = 0
    int c1 = 0, r1 = 1 + 2 * half;      // v = 1
    const float* w0k = w0;              // advances 4 rows per k-step (uniform)
    for (int k0 = 0; k0 < 576; k0 += 4) {
        v2f a;
        {   // v = 0
            const int di = (r0 * 11) >> 5;        // r/3 for r in [0,8]
            const int dj = r0 - di * 3;
            const int yo = (di == 1) ? yoff1 : ((di == 2) ? yoff2 : yoff0);
            const int xo = (dj == 1) ? xoff1 : ((dj == 2) ? xoff2 : xoff0);
            const int ok = (vy >> di) & (vx >> dj) & 1;
            const float val = featb[c0 * (H_ * W_) + yo + xo];
            a[0] = ok ? val : 0.0f;
            const int rn = r0 + 4;
            const int wrap = rn >= 9;
            r0 = wrap ? rn - 9 : rn;
            c0 += wrap ? 1 : 0;
        }
        {   // v = 1
            const int di = (r1 * 11) >> 5;
            const int dj = r1 - di * 3;
            const int yo = (di == 1) ? yoff1 : ((di == 2) ? yoff2 : yoff0);
            const int xo = (dj == 1) ? xoff1 : ((dj == 2) ? xoff2 : xoff0);
            const int ok = (vy >> di) & (vx >> dj) & 1;
            const float val = featb[c1 * (H_ * W_) + yo + xo];
            a[1] = ok ? val : 0.0f;
            const int rn = r1 + 4;
            const int wrap = rn >= 9;
            r1 = wrap ? rn - 9 : rn;
            c1 += wrap ? 1 : 0;
        }
#pragma unroll
        for (int nt = 0; nt < 16; ++nt) {
            v2f bb;
            bb[0] = w0k[(2 * half) * NHID + nt * 16 + nl];
            bb[1] = w0k[(1 + 2 * half) * NHID + nt * 16 + nl];
            acc[nt] = __builtin_amdgcn_wmma_f32_16x16x4_f32(
                false, a, false, bb, (short)0, acc[nt], false, false);
        }
        w0k += 4 * NHID;
    }
    // activation -> LDS (A layout for next layer), with the pe column folded in
    // as a rank-1 FMA: pre_act(m, n) += pe_m * w0[576, n]. Same-wave LDS is
    // in-order, so no barrier is needed.
#pragma unroll
    for (int nt = 0; nt < 16; ++nt) {
        const int n = nt * 16 + nl;
        const float bias = b0[n];
        const float w0c  = w0[576 * NHID + n];
#pragma unroll
        for (int v = 0; v < 8; ++v)
            hl[v + 8 * half][n] = __sinf(W0S * (acc[nt][v] + qpe[v] * w0c + bias));
    }

    // =======================  layers 1..3: 256 x 256  ============================
    const float* Ws[3]  = {w1, w2, w3};
    const float* Bss[3] = {b1, b2, b3};
    for (int L = 0; L < 3; ++L) {
        const float* wk = Ws[L];     // uniform pointer, advances 4 rows / k-step
#pragma unroll
        for (int t = 0; t < 16; ++t) acc[t] = zero8;
        const float* arow = &hl[nl][2 * half];
        for (int k0 = 0; k0 < NHID; k0 += 4) {
            v2f a;
            a[0] = arow[k0];
            a[1] = arow[k0 + 1];
#pragma unroll
            for (int nt = 0; nt < 16; ++nt) {
                v2f bb;
                bb[0] = wk[(2 * half) * NHID + nt * 16 + nl];
                bb[1] = wk[(1 + 2 * half) * NHID + nt * 16 + nl];
                acc[nt] = __builtin_amdgcn_wmma_f32_16x16x4_f32(
                    false, a, false, bb, (short)0, acc[nt], false, false);
            }
            wk += 4 * NHID;
        }
        const float* bias_p = Bss[L];
#pragma unroll
        for (int nt = 0; nt < 16; ++nt) {
            const int n = nt * 16 + nl;
            const float bias = bias_p[n];
#pragma unroll
            for (int v = 0; v < 8; ++v)
                hl[v + 8 * half][n] = __sinf(W0S * (acc[nt][v] + bias));
        }
    }

    // =======================  layer 4: 256 -> 27 (pad N to 32)  ==================
    // Tile 1 columns are clamped to 26: junk-but-finite values in columns 27..31
    // are never stored (fold3 scatter skips n >= 27).
    v8f oacc[2];
    oacc[0] = zero8; oacc[1] = zero8;
    {
        const int n1c = min(16 + nl, NOUT - 1);
        const float* wk = w4;
        const float* arow = &hl[nl][2 * half];
        for (int k0 = 0; k0 < NHID; k0 += 4) {
            v2f a;
            a[0] = arow[k0];
            a[1] = arow[k0 + 1];
            v2f bb0, bb1;
            bb0[0] = wk[(2 * half) * NOUT + nl];
            bb0[1] = wk[(1 + 2 * half) * NOUT + nl];
            bb1[0] = wk[(2 * half) * NOUT + n1c];
            bb1[1] = wk[(1 + 2 * half) * NOUT + n1c];
            oacc[0] = __builtin_amdgcn_wmma_f32_16x16x4_f32(
                false, a, false, bb0, (short)0, oacc[0], false, false);
            oacc[1] = __builtin_amdgcn_wmma_f32_16x16x4_f32(
                false, a, false, bb1, (short)0, oacc[1], false, false);
            wk += 4 * NOUT;
        }
    }

    // =======================  fold3 scatter-add  =================================
    // pat channel n = co*9 + i*3 + j contributes to out[b, co, iy+i-1, ix+j-1].
#pragma unroll
    for (int nt = 0; nt < 2; ++nt) {
        const int n = nt * 16 + nl;
        if (n < NOUT) {
            const float bias = b4[n];
            const int co = n / 9;
            const int r  = n - co * 9;
            const int di = r / 3 - 1;
            const int dj = (r - (r / 3) * 3) - 1;
#pragma unroll
            for (int v = 0; v < 8; ++v) {
                const float val = oacc[nt][v] + bias;
                const int y = qy[v] + di;
                const int x = qx[v] + dj;
                if (y >= 0 && y < H_ && x >= 0 && x < W_)
                    atomicAdd(&out[((qb[v] * 3 + co) * H_ + y) * W_ + x], val);
            }
        }
    }
}

extern "C" void kernel_launch(void* const* d_in, const int* in_sizes, int n_in,
                              void* d_out, int out_size, void* d_ws, size_t ws_size,
                              hipStream_t stream) {
    const float* feat  = (const float*)d_in[0];
    const float* coord = (const float*)d_in[1];
    const float* w0 = (const float*)d_in[2];
    const float* b0 = (const float*)d_in[3];
    const float* w1 = (const float*)d_in[4];
    const float* b1 = (const float*)d_in[5];
    const float* w2 = (const float*)d_in[6];
    const float* b2 = (const float*)d_in[7];
    const float* w3 = (const float*)d_in[8];
    const float* b3 = (const float*)d_in[9];
    const float* w4 = (const float*)d_in[10];
    const float* b4 = (const float*)d_in[11];
    float* out = (float*)d_out;

    // fold3 output accumulated with atomics -> must start from zero.
    zero_f32_kernel<<<(out_size + 255) / 256, 256, 0, stream>>>(out, out_size);

    const int nwaves  = (NB * Q_) / 16;  // 8192 wave tiles
    const int nblocks = nwaves / 2;      // 2 waves per block
    liif_mlp_kernel<<<nblocks, 64, 0, stream>>>(
        feat, coord, w0, b0, w1, b1, w2, b2, w3, b3, w4, b4, out);
}